// SAGE_17703855194786
// MI455X (gfx1250) — compile-verified
//
#include <hip/hip_runtime.h>
#include <hip/hip_bf16.h>
#include <stdint.h>

#define N_NODES 100000
#define N_EDGES 1600000
#define FEAT    128
#define N_CLS   40

typedef __attribute__((ext_vector_type(2))) float v2f;
typedef __attribute__((ext_vector_type(8))) float v8f;

// V_WMMA_F32_16X16X4_F32 accumulate step.
// A fragment (16x4 f32): lanes 0-15 hold K=k0,k0+1; lanes 16-31 hold K=k0+2,k0+3
// for row M = lane&15.  B fragment (4x16) mirrors it for col N = lane&15.
__device__ __forceinline__ v8f wmma4(v2f a, v2f b, v8f c)
{
    return __builtin_amdgcn_wmma_f32_16x16x4_f32(
        /*neg_a=*/false, a, /*neg_b=*/false, b,
        /*c_mod=*/(short)0, c, /*reuse_a=*/false, /*reuse_b=*/false);
}

// ---------------------------------------------------------------------------
// deg[i] = number of incoming edges (float, for the mean divide)
// ---------------------------------------------------------------------------
__global__ __launch_bounds__(256) void deg_kernel(const long long* __restrict__ dst,
                                                  float* __restrict__ deg)
{
    int e = blockIdx.x * 256 + threadIdx.x;
    if (e < N_EDGES) atomicAdd(&deg[(int)dst[e]], 1.0f);
}

// ---------------------------------------------------------------------------
// agg[dst] += feat[src]  (one thread per (edge, 4-float group); a wave covers
// one edge's full 512B row coalesced; atomics land in the 192MB L2)
// ---------------------------------------------------------------------------
__global__ __launch_bounds__(256) void scatter_kernel(const float* __restrict__ feat,
                                                      const long long* __restrict__ src,
                                                      const long long* __restrict__ dst,
                                                      float* __restrict__ agg)
{
    long long t = (long long)blockIdx.x * 256 + threadIdx.x;
    int e = (int)(t >> 5);
    int g = (int)(t & 31);
    if (e >= N_EDGES) return;
    int s = (int)src[e];
    int d = (int)dst[e];
    const float4 v = *reinterpret_cast<const float4*>(feat + (size_t)s * FEAT + g * 4);
    float* p = agg + (size_t)d * FEAT + g * 4;
    atomicAdd(p + 0, v.x);
    atomicAdd(p + 1, v.y);
    atomicAdd(p + 2, v.z);
    atomicAdd(p + 3, v.w);
}

// ---------------------------------------------------------------------------
// out = relu( (agg/deg) @ Wl^T + bl + xin @ Wr^T )  — fused dual GEMM.
// One wave owns a full 16x128 output strip (8 column tiles, C in 64 VGPRs):
// each A fragment (mean + root) is loaded ONCE per k-step and reused across
// all 8 column tiles -> 16 WMMAs per 2 A loads.  Weight fragments come from
// one per-lane base pointer + immediate offsets (t*8192 B), cache-resident.
// Block = 64 threads = 2 waves = 2 row tiles; grid = N/32 = 3125 (exact).
// ---------------------------------------------------------------------------
__global__ __launch_bounds__(64) void sage_layer_kernel(const float* __restrict__ xin,
                                                        const float* __restrict__ agg,
                                                        const float* __restrict__ deg,
                                                        const float* __restrict__ Wl,
                                                        const float* __restrict__ bl,
                                                        const float* __restrict__ Wr,
                                                        float* __restrict__ out)
{
    const int lane    = threadIdx.x & 31;
    const int wave    = threadIdx.x >> 5;       // 0..1 -> row tile within block
    const int half    = lane >> 4;              // 0/1
    const int l16     = lane & 15;
    const int rowTile = blockIdx.x * 2 + wave;  // 0..6249
    const int row     = rowTile * 16 + l16;     // A-fragment row for this lane

    const float inv = 1.0f / fmaxf(deg[row], 1.0f);  // mean = agg/max(deg,1)

    const float* Am  = agg + (size_t)row * FEAT + 2 * half;  // mean operand
    const float* Ax  = xin + (size_t)row * FEAT + 2 * half;  // root operand
    const float* Bl0 = Wl  + (size_t)l16 * FEAT + 2 * half;  // B[k][o] = W[o][k]
    const float* Br0 = Wr  + (size_t)l16 * FEAT + 2 * half;

    v8f c[8] = {{}, {}, {}, {}, {}, {}, {}, {}};

#pragma unroll
    for (int k = 0; k < FEAT; k += 4) {
        v2f aM, aX;
        aM.x = Am[k] * inv;
        aM.y = Am[k + 1] * inv;
        aX.x = Ax[k];
        aX.y = Ax[k + 1];
#pragma unroll
        for (int t = 0; t < 8; ++t) {
            v2f b;
            b.x  = Bl0[t * 2048 + k];       // t*16 rows * 128 floats
            b.y  = Bl0[t * 2048 + k + 1];
            c[t] = wmma4(aM, b, c[t]);      // mean @ Wl^T
            b.x  = Br0[t * 2048 + k];
            b.y  = Br0[t * 2048 + k + 1];
            c[t] = wmma4(aX, b, c[t]);      // x @ Wr^T
        }
    }

    const int orow0 = rowTile * 16 + half * 8;  // C: VGPR r <-> M = r + 8*half
#pragma unroll
    for (int t = 0; t < 8; ++t) {
        const int col  = t * 16 + l16;
        const float bias = bl[col];
#pragma unroll
        for (int r = 0; r < 8; ++r) {
            out[(size_t)(orow0 + r) * FEAT + col] = fmaxf(c[t][r] + bias, 0.0f);
        }
    }
}

// ---------------------------------------------------------------------------
// out = h @ Wc^T + bc   (N x 40); one wave owns 16 rows x 3 column tiles.
// Tile 2 covers cols 32..47; cols >= 40 are masked arithmetically (selects,
// no EXEC change) before the WMMAs, and the store is guarded after.
// ---------------------------------------------------------------------------
__global__ __launch_bounds__(64) void classifier_kernel(const float* __restrict__ h,
                                                        const float* __restrict__ Wc,
                                                        const float* __restrict__ bc,
                                                        float* __restrict__ out)
{
    const int lane    = threadIdx.x & 31;
    const int wave    = threadIdx.x >> 5;
    const int half    = lane >> 4;
    const int l16     = lane & 15;
    const int rowTile = blockIdx.x * 2 + wave;
    const int row     = rowTile * 16 + l16;

    const float* Arow = h  + (size_t)row * FEAT + 2 * half;
    const float* B0   = Wc + (size_t)l16 * FEAT + 2 * half;   // tiles 0,1: +0, +2048

    // tile 2: cols 32 + l16, valid only for l16 < 8 (cols 32..39)
    const int   l16c = (l16 < 8) ? l16 : 7;
    const float m2   = (l16 < 8) ? 1.0f : 0.0f;
    const float* B2  = Wc + (size_t)(32 + l16c) * FEAT + 2 * half;

    v8f c[3] = {{}, {}, {}};

#pragma unroll
    for (int k = 0; k < FEAT; k += 4) {
        v2f a, b;
        a.x = Arow[k];
        a.y = Arow[k + 1];

        b.x = B0[k];
        b.y = B0[k + 1];
        c[0] = wmma4(a, b, c[0]);

        b.x = B0[2048 + k];
        b.y = B0[2048 + k + 1];
        c[1] = wmma4(a, b, c[1]);

        b.x = B2[k] * m2;
        b.y = B2[k + 1] * m2;
        c[2] = wmma4(a, b, c[2]);
    }

    const int orow0 = rowTile * 16 + half * 8;
#pragma unroll
    for (int t = 0; t < 3; ++t) {
        const int col = t * 16 + l16;
        if (col < N_CLS) {
            const float bias = bc[col];
#pragma unroll
            for (int r = 0; r < 8; ++r) {
                out[(size_t)(orow0 + r) * N_CLS + col] = c[t][r] + bias;
            }
        }
    }
}

// ---------------------------------------------------------------------------
extern "C" void kernel_launch(void* const* d_in, const int* in_sizes, int n_in,
                              void* d_out, int out_size, void* d_ws, size_t ws_size,
                              hipStream_t stream)
{
    const float*     x    = (const float*)d_in[0];
    const long long* ei   = (const long long*)d_in[1];  // int64 per reference
    /* d_in[2] edge_weight: unused by the reference computation */
    const float*     W1l  = (const float*)d_in[3];
    const float*     b1l  = (const float*)d_in[4];
    const float*     W1r  = (const float*)d_in[5];
    const float*     W2l  = (const float*)d_in[6];
    const float*     b2l  = (const float*)d_in[7];
    const float*     W2r  = (const float*)d_in[8];
    const float*     Wc   = (const float*)d_in[9];
    const float*     bc   = (const float*)d_in[10];

    const long long* src = ei;             // edge_index[0]
    const long long* dst = ei + N_EDGES;   // edge_index[1]

    // Workspace layout (floats): deg | agg | h1 | h2  (~154 MB total)
    float* deg = (float*)d_ws;
    float* agg = deg + N_NODES;
    float* h1  = agg + (size_t)N_NODES * FEAT;
    float* h2  = h1  + (size_t)N_NODES * FEAT;

    const size_t aggBytes = (size_t)N_NODES * FEAT * sizeof(float);
    const long long scatterThreads = (long long)N_EDGES * 32;
    const int scatterBlocks = (int)((scatterThreads + 255) / 256);
    const int gemmBlocks = N_NODES / 32;   // 3125, exact (2 row tiles / block)

    // degrees (shared by both layers)
    hipMemsetAsync(deg, 0, N_NODES * sizeof(float), stream);
    deg_kernel<<<(N_EDGES + 255) / 256, 256, 0, stream>>>(dst, deg);

    // ---- layer 1 ----
    hipMemsetAsync(agg, 0, aggBytes, stream);
    scatter_kernel<<<scatterBlocks, 256, 0, stream>>>(x, src, dst, agg);
    sage_layer_kernel<<<gemmBlocks, 64, 0, stream>>>(x, agg, deg, W1l, b1l, W1r, h1);

    // ---- layer 2 ----
    hipMemsetAsync(agg, 0, aggBytes, stream);
    scatter_kernel<<<scatterBlocks, 256, 0, stream>>>(h1, src, dst, agg);
    sage_layer_kernel<<<gemmBlocks, 64, 0, stream>>>(h1, agg, deg, W2l, b2l, W2r, h2);

    // ---- classifier ----
    classifier_kernel<<<gemmBlocks, 64, 0, stream>>>(h2, Wc, bc, (float*)d_out);
}